// _GCNConvFC_23510650978598
// MI455X (gfx1250) — compile-verified
//
#include <hip/hip_runtime.h>
#include <hip/hip_bf16.h>
#include <math.h>

typedef __attribute__((ext_vector_type(2))) float v2f;
typedef __attribute__((ext_vector_type(4))) float v4f;
typedef __attribute__((ext_vector_type(8))) float v8f;

// 16-byte int vector matching the builtin's parameter type
// (clang printed: "__attribute__((__vector_size__(4 * sizeof(int)))) int __device__ *")
typedef int v4i_vs __attribute__((vector_size(16)));
typedef __attribute__((address_space(1))) v4i_vs gas_v4i;   // global (AS1)
typedef __attribute__((address_space(3))) v4i_vs las_v4i;   // LDS    (AS3)

// ---------------- async global->LDS staging helpers ----------------
// CDNA5 GLOBAL_LOAD_ASYNC_TO_LDS_B128 (ASYNCcnt-tracked); sync float4 copy
// fallback if the builtin is unavailable.

#if defined(__gfx1250__) && __has_builtin(__builtin_amdgcn_global_load_async_to_lds_b128)
#define ATHENA_HAS_ASYNC_LDS 1
#else
#define ATHENA_HAS_ASYNC_LDS 0
#endif

__device__ __forceinline__ void stage_b128(const float* __restrict__ g,
                                           float* __restrict__ l) {
#if ATHENA_HAS_ASYNC_LDS
    float* gnc = const_cast<float*>(g);
    __builtin_amdgcn_global_load_async_to_lds_b128(
        (gas_v4i*)gnc, (las_v4i*)l, 0, 0);
#else
    *(v4f*)l = *(const v4f*)g;
#endif
}

__device__ __forceinline__ void wait_async_copies() {
#if ATHENA_HAS_ASYNC_LDS
#if __has_builtin(__builtin_amdgcn_s_wait_asynccnt)
    __builtin_amdgcn_s_wait_asynccnt(0);
#else
    asm volatile("s_wait_asynccnt 0" ::: "memory");
#endif
#endif
}

// ---------------- small utility kernels ----------------

__global__ void k_fill_f32(float* __restrict__ p, float v, int n) {
    int i = blockIdx.x * blockDim.x + threadIdx.x;
    if (i < n) p[i] = v;
}

// deg[dst] += 1 for every edge (self-loop contribution pre-filled with 1.0)
__global__ void k_count_deg(const int* __restrict__ ei, float* __restrict__ deg,
                            int E) {
    int e = blockIdx.x * blockDim.x + threadIdx.x;
    if (e < E) {
        int d = ei[E + e];
        atomicAdd(&deg[d], 1.0f);
    }
}

__global__ void k_rsqrt_inplace(float* __restrict__ p, int n) {
    int i = blockIdx.x * blockDim.x + threadIdx.x;
    if (i < n) p[i] = rsqrtf(p[i]);   // deg >= 1 always (self loops)
}

// ---------------- GEMM: XW = X @ W  via V_WMMA_F32_16X16X4_F32 ----------------
// Workgroup = 4 waves. Each wave computes one 16(M) x 64(N) strip; the 4 waves
// share a KCx64 B panel staged in LDS (async global->LDS), cutting W re-fetch
// traffic 4x. K loop steps by 4 (f32 WMMA depth).
#define GKC 64   // K-chunk staged in LDS: GKC x 64 floats = 16 KB

__global__ __launch_bounds__(128)
void k_gemm_wmma(const float* __restrict__ X, const float* __restrict__ W,
                 float* __restrict__ XW, int D, int mtiles) {
    __shared__ float Bs[GKC * 64];

    const int tid  = threadIdx.x;
    const int wave = tid >> 5;
    const int lane = tid & 31;
    const int half = lane >> 4;     // 0: lanes 0-15, 1: lanes 16-31
    const int l16  = lane & 15;
    const int tile = blockIdx.x * 4 + wave;       // per-wave M tile
    const bool valid = (tile < mtiles);           // uniform across the wave
    const int row  = tile * 16 + l16;             // A-matrix row (M index)
    const int nc0  = blockIdx.y * 64;             // first output column

    v8f c0 = {}, c1 = {}, c2 = {}, c3 = {};

    for (int kc = 0; kc < D; kc += GKC) {
        __syncthreads();   // previous chunk's readers done before overwrite
        // cooperative stage of B panel: GKC x 64 floats, 8 x b128 per thread
        for (int i = 0; i < (GKC * 64 / 4) / 128; ++i) {
            int f4 = tid + i * 128;       // float4 index within panel
            int r  = f4 >> 4;             // 16 float4 per 64-wide row
            int c  = (f4 & 15) << 2;
            stage_b128(W + (size_t)(kc + r) * D + nc0 + c, &Bs[r * 64 + c]);
        }
        wait_async_copies();
        __syncthreads();

        if (valid) {
            for (int k0 = 0; k0 < GKC; k0 += 4) {
                // A 16x4: lanes 0-15 -> K=k0,k0+1 ; lanes 16-31 -> K=k0+2,k0+3
                v2f a = *(const v2f*)(X + (size_t)row * D + kc + k0 + half * 2);
                // B 4x16 fragments from LDS panel
                const float* br0 = &Bs[(k0 + half) * 64 + l16];
                const float* br1 = &Bs[(k0 + 2 + half) * 64 + l16];
                v2f b0, b1, b2, b3;
                b0.x = br0[0];  b0.y = br1[0];
                b1.x = br0[16]; b1.y = br1[16];
                b2.x = br0[32]; b2.y = br1[32];
                b3.x = br0[48]; b3.y = br1[48];

                c0 = __builtin_amdgcn_wmma_f32_16x16x4_f32(false, a, false, b0, (short)0, c0, false, false);
                c1 = __builtin_amdgcn_wmma_f32_16x16x4_f32(false, a, false, b1, (short)0, c1, false, false);
                c2 = __builtin_amdgcn_wmma_f32_16x16x4_f32(false, a, false, b2, (short)0, c2, false, false);
                c3 = __builtin_amdgcn_wmma_f32_16x16x4_f32(false, a, false, b3, (short)0, c3, false, false);
            }
        }
    }

    if (valid) {
        // C/D layout: VGPR r holds row (r + half*8), col = l16 within tile
        const int orow = tile * 16 + half * 8;
        for (int r = 0; r < 8; ++r) {
            float* out = XW + (size_t)(orow + r) * D + nc0 + l16;
            out[0]  = c0[r];
            out[16] = c1[r];
            out[32] = c2[r];
            out[48] = c3[r];
        }
    }
}

// Scalar fallback for M-remainder rows (unused for N=50000 but kept general).
__global__ void k_gemm_rows_scalar(const float* __restrict__ X,
                                   const float* __restrict__ W,
                                   float* __restrict__ XW,
                                   int row0, int nrows, int D) {
    int r = row0 + blockIdx.x;
    if (blockIdx.x >= nrows) return;
    int c = threadIdx.x;                 // blockDim.x == D
    float acc = 0.f;
    for (int k = 0; k < D; ++k) acc += X[(size_t)r * D + k] * W[(size_t)k * D + c];
    XW[(size_t)r * D + c] = acc;
}

// ---------------- aggregation ----------------

// agg[n] = b + xw[n] * dinv[n]^2   (bias + self-loop message)
__global__ void k_agg_init(const float* __restrict__ xw, const float* __restrict__ dinv,
                           const float* __restrict__ b, float* __restrict__ agg, int D) {
    int n = blockIdx.x;
    int t = threadIdx.x;                 // blockDim.x == D
    float di = dinv[n];
    agg[(size_t)n * D + t] = b[t] + xw[(size_t)n * D + t] * di * di;
}

// agg[dst] += xw[src] * dinv[src] * dinv[dst]   (one block per edge, coalesced row)
__global__ void k_scatter_edges(const int* __restrict__ ei,
                                const float* __restrict__ xw,
                                const float* __restrict__ dinv,
                                float* __restrict__ agg, int E, int D) {
    int e = blockIdx.x;
    int t = threadIdx.x;                 // blockDim.x == D
    int s = ei[e];
    int d = ei[E + e];
    float nrm = dinv[s] * dinv[d];
    float v = xw[(size_t)s * D + t] * nrm;
    atomicAdd(&agg[(size_t)d * D + t], v);
}

// ---------------- BatchNorm statistics ----------------

// per-feature partial sums over rows, fully coalesced, atomic-reduced
__global__ void k_bn_stats(const float* __restrict__ agg, float* __restrict__ sum,
                           float* __restrict__ sumsq, int N, int D) {
    int t = threadIdx.x;                 // blockDim.x == D
    float s = 0.f, s2 = 0.f;
    for (int r = blockIdx.x; r < N; r += gridDim.x) {
        float v = agg[(size_t)r * D + t];
        s += v;
        s2 += v * v;
    }
    atomicAdd(&sum[t], s);
    atomicAdd(&sumsq[t], s2);
}

__global__ void k_bn_finalize(const float* __restrict__ sum, const float* __restrict__ sumsq,
                              const float* __restrict__ gamma, const float* __restrict__ beta,
                              float* __restrict__ scale, float* __restrict__ shift,
                              int N, float eps) {
    int t = threadIdx.x;                 // one block, blockDim.x == D
    float invN = 1.0f / (float)N;
    float mu = sum[t] * invN;
    float var = sumsq[t] * invN - mu * mu;
    float sc = gamma[t] * rsqrtf(var + eps);
    scale[t] = sc;
    shift[t] = beta[t] - mu * sc;
}

// ---------------- BN apply + ReLU + global mean pool (scatter) ----------------

__global__ void k_bn_relu_pool(const float* __restrict__ agg, const int* __restrict__ batch,
                               const float* __restrict__ scale, const float* __restrict__ shift,
                               float* __restrict__ psum, float* __restrict__ cnt, int D) {
    int n = blockIdx.x;
    int t = threadIdx.x;                 // blockDim.x == D
    int g = batch[n];
    float h = fmaxf(scale[t] * agg[(size_t)n * D + t] + shift[t], 0.0f);
    atomicAdd(&psum[(size_t)g * D + t], h);
    if (t == 0) atomicAdd(&cnt[g], 1.0f);
}

// ---------------- final FC + log_softmax (one block per graph) ----------------

__global__ void k_fc_logsoftmax(const float* __restrict__ psum, const float* __restrict__ cnt,
                                const float* __restrict__ Wf, const float* __restrict__ bf,
                                float* __restrict__ out, int D, int C) {
    extern __shared__ float sh[];        // D (pooled) + C (logits) + 1 (lse)
    float* pooled = sh;
    float* logits = sh + D;
    int g = blockIdx.x;
    float inv = 1.0f / fmaxf(cnt[g], 1.0f);
    for (int t = threadIdx.x; t < D; t += blockDim.x)
        pooled[t] = psum[(size_t)g * D + t] * inv;
    __syncthreads();
    for (int c = threadIdx.x; c < C; c += blockDim.x) {
        float acc = bf[c];
        const float* w = Wf + (size_t)c * D;
        for (int k = 0; k < D; ++k) acc += pooled[k] * w[k];
        logits[c] = acc;
    }
    __syncthreads();
    if (threadIdx.x == 0) {
        float m = -INFINITY;
        for (int i = 0; i < C; ++i) m = fmaxf(m, logits[i]);
        float s = 0.f;
        for (int i = 0; i < C; ++i) s += expf(logits[i] - m);
        sh[D + C] = m + logf(s);
    }
    __syncthreads();
    float lse = sh[D + C];
    for (int c = threadIdx.x; c < C; c += blockDim.x)
        out[(size_t)g * C + c] = logits[c] - lse;
}

// ---------------- launcher ----------------

extern "C" void kernel_launch(void* const* d_in, const int* in_sizes, int n_in,
                              void* d_out, int out_size, void* d_ws, size_t ws_size,
                              hipStream_t stream) {
    const float* x     = (const float*)d_in[0];
    const int*   ei    = (const int*)  d_in[1];   // [2, E] flattened
    const int*   batch = (const int*)  d_in[2];
    const float* W     = (const float*)d_in[3];
    const float* b     = (const float*)d_in[4];
    const float* gamma = (const float*)d_in[5];
    const float* beta  = (const float*)d_in[6];
    const float* Wf    = (const float*)d_in[7];
    const float* bf    = (const float*)d_in[8];
    float* out = (float*)d_out;

    const int N = in_sizes[2];
    const int E = in_sizes[1] / 2;
    const int D = in_sizes[4];            // 256
    const int C = in_sizes[8];            // 60
    const int G = out_size / C;           // 64

    // workspace carve-out (256B aligned)
    char* ws = (char*)d_ws;
    size_t off = 0;
    auto carve = [&](size_t nfloat) -> float* {
        float* p = (float*)(ws + off);
        off = (off + nfloat * sizeof(float) + 255) & ~(size_t)255;
        return p;
    };
    float* xw    = carve((size_t)N * D);
    float* agg   = carve((size_t)N * D);
    float* dinv  = carve(N);              // holds deg first, then rsqrt in place
    float* stats = carve(2 * D);          // sum | sumsq
    float* scale = carve(D);
    float* shift = carve(D);
    float* psum  = carve((size_t)G * D + G); // pooled sums | counts
    float* cnt   = psum + (size_t)G * D;
    (void)ws_size;

    // 1) degree (self loop = 1) and dinv
    k_fill_f32<<<(N + 255) / 256, 256, 0, stream>>>(dinv, 1.0f, N);
    k_count_deg<<<(E + 255) / 256, 256, 0, stream>>>(ei, dinv, E);
    k_rsqrt_inplace<<<(N + 255) / 256, 256, 0, stream>>>(dinv, N);

    // 2) xw = x @ W  (WMMA f32 16x16x4; 4 waves share an LDS-staged B panel)
    {
        int mtiles = N / 16;
        if (mtiles > 0) {
            dim3 grid((mtiles + 3) / 4, D / 64);
            k_gemm_wmma<<<grid, 128, 0, stream>>>(x, W, xw, D, mtiles);
        }
        int rem = N - mtiles * 16;
        if (rem > 0)
            k_gemm_rows_scalar<<<rem, D, 0, stream>>>(x, W, xw, mtiles * 16, rem, D);
    }

    // 3) agg = b + self-loop message; then edge scatter-add
    k_agg_init<<<N, D, 0, stream>>>(xw, dinv, b, agg, D);
    k_scatter_edges<<<E, D, 0, stream>>>(ei, xw, dinv, agg, E, D);

    // 4) BatchNorm stats -> scale/shift
    k_fill_f32<<<(2 * D + 255) / 256, 256, 0, stream>>>(stats, 0.0f, 2 * D);
    k_bn_stats<<<512, D, 0, stream>>>(agg, stats, stats + D, N, D);
    k_bn_finalize<<<1, D, 0, stream>>>(stats, stats + D, gamma, beta, scale, shift, N, 1e-5f);

    // 5) BN+ReLU fused with global mean-pool scatter
    k_fill_f32<<<((G * D + G) + 255) / 256, 256, 0, stream>>>(psum, 0.0f, G * D + G);
    k_bn_relu_pool<<<N, D, 0, stream>>>(agg, batch, scale, shift, psum, cnt, D);

    // 6) FC + log_softmax
    size_t shbytes = (size_t)(D + C + 1) * sizeof(float);
    k_fc_logsoftmax<<<G, 64, shbytes, stream>>>(psum, cnt, Wf, bf, out, D, C);
}